// JavascriptExtractor_33260226740802
// MI455X (gfx1250) — compile-verified
//
#include <hip/hip_runtime.h>

// Problem shape (fixed by the reference)
#define S_SRC 2048
#define S_Q   512
#define OUT_L (S_SRC + S_Q)   // 2560
#define NTHREADS 256

typedef int v4i __attribute__((ext_vector_type(4)));

// One workgroup per row:
//  1) async-copy the row of sources (8 KB) and queries (2 KB) into LDS using
//     gfx1250 global_load_async_to_lds_b128 (ASYNCcnt path), GVS saddr form,
//     non-temporal (streamed once, reused only from LDS)
//  2) count nonzeros (front-packed rows => count == length) with a wave32
//     shuffle reduction + LDS atomics
//  3) write the shifted concat with non-temporal b128 stores, branchless selects
__global__ __launch_bounds__(NTHREADS) void ragged_concat_kernel(
    const int* __restrict__ src, const int* __restrict__ qry,
    int* __restrict__ out)
{
    __shared__ __align__(16) int s_src[S_SRC];
    __shared__ __align__(16) int s_q[S_Q];
    __shared__ int s_cnt[2];

    const int tid = threadIdx.x;
    const long long row = blockIdx.x;

    if (tid < 2) s_cnt[tid] = 0;

    // Uniform 64-bit row bases (SGPRs via saddr form)
    unsigned long long gsrc = (unsigned long long)(const void*)(src + row * S_SRC);
    unsigned long long gq   = (unsigned long long)(const void*)(qry + row * S_Q);
    // LDS byte offsets (low 32 bits of the flat pointer are the LDS offset)
    unsigned lsrc = (unsigned)(size_t)s_src;
    unsigned lq   = (unsigned)(size_t)s_q;

    // ---- Phase 1: async global -> LDS, 16 B per lane-op, GVS addressing ----
    // sources: 8192 B = 512 x b128 ops -> 2 per thread
    // queries: 2048 B = 128 x b128 ops -> first 128 threads
    unsigned o = (unsigned)tid * 16u;
    asm volatile("global_load_async_to_lds_b128 %0, %1, %2 th:TH_LOAD_NT"
                 :: "v"(lsrc + o), "v"(o), "s"(gsrc) : "memory");
    asm volatile("global_load_async_to_lds_b128 %0, %1, %2 th:TH_LOAD_NT"
                 :: "v"(lsrc + o + 4096u), "v"(o + 4096u), "s"(gsrc) : "memory");
    if (tid < (S_Q * 4 / 16)) {
        asm volatile("global_load_async_to_lds_b128 %0, %1, %2 th:TH_LOAD_NT"
                     :: "v"(lq + o), "v"(o), "s"(gq) : "memory");
    }
    // Drain this wave's async copies, then barrier so every wave sees all LDS data
    asm volatile("s_wait_asynccnt 0x0" ::: "memory");
    __syncthreads();

    // ---- Phase 2: per-row lengths ----
    int c = 0;
#pragma unroll
    for (int i = 0; i < S_SRC / NTHREADS; ++i)
        c += (s_src[tid + i * NTHREADS] > 0) ? 1 : 0;
    int cq = 0;
#pragma unroll
    for (int i = 0; i < S_Q / NTHREADS; ++i)
        cq += (s_q[tid + i * NTHREADS] > 0) ? 1 : 0;

    // wave32 tree reduction (gfx1250 is wave32-only)
#pragma unroll
    for (int off = 16; off > 0; off >>= 1) {
        c  += __shfl_down(c, off, 32);
        cq += __shfl_down(cq, off, 32);
    }
    if ((tid & 31) == 0) {
        atomicAdd(&s_cnt[0], c);
        atomicAdd(&s_cnt[1], cq);
    }
    __syncthreads();

    const int sl = s_cnt[0];
    const int e  = sl + s_cnt[1];
    int* orow = out + row * (long long)OUT_L;

    // ---- Phase 3: shifted concat, non-temporal b128 stores ----
    for (int v = tid; v < OUT_L / 4; v += NTHREADS) {
        int base = v * 4;
        int tmp[4];
#pragma unroll
        for (int j = 0; j < 4; ++j) {
            int pos = base + j;
            int qi  = pos - sl;
            qi = qi < 0 ? 0 : (qi > S_Q - 1 ? S_Q - 1 : qi);
            int x = (pos < sl) ? s_src[pos] : s_q[qi];
            tmp[j] = (pos < e) ? x : 0;
        }
        v4i val = { tmp[0], tmp[1], tmp[2], tmp[3] };
        __builtin_nontemporal_store(val, (v4i*)(orow + base));
    }
}

extern "C" void kernel_launch(void* const* d_in, const int* in_sizes, int n_in,
                              void* d_out, int out_size, void* d_ws, size_t ws_size,
                              hipStream_t stream) {
    const int* sources = (const int*)d_in[0];   // [N, 2048] int32
    const int* queries = (const int*)d_in[1];   // [N, 512]  int32
    // d_in[2] (targets) is unused by the reference
    int* out = (int*)d_out;                     // [N, 2560] int32

    const int n = in_sizes[0] / S_SRC;          // 16384 rows
    ragged_concat_kernel<<<n, NTHREADS, 0, stream>>>(sources, queries, out);
}